// GATLayer_49400713839387
// MI455X (gfx1250) — compile-verified
//
#include <hip/hip_runtime.h>
#include <math.h>

// Problem constants (match reference)
#define N_NODES 8192
#define IN_DIM  128
#define OUT_DIM 64
#define N_EDGES 262144
#define NEG_SLOPE 0.2f

// Workspace layout (bytes):
//   h      : N_NODES * OUT_DIM * 4      = 2,097,152
//   left   : N_NODES * 4                =    32,768
//   right  : N_NODES * 4                =    32,768
//   bitmap : N_NODES * (N_NODES/32) * 4 = 8,388,608   (adjacency, dedupes edges)
#define WS_H_OFF      0
#define WS_LEFT_OFF   (N_NODES * OUT_DIM * 4)
#define WS_RIGHT_OFF  (WS_LEFT_OFF + N_NODES * 4)
#define WS_BMP_OFF    (WS_RIGHT_OFF + N_NODES * 4)
#define BMP_WORDS_PER_ROW (N_NODES / 32)   // 256
#define BMP_TOTAL_WORDS   (N_NODES * BMP_WORDS_PER_ROW)  // 2,097,152

typedef float v2f __attribute__((ext_vector_type(2)));
typedef float v8f __attribute__((ext_vector_type(8)));

// ---------------------------------------------------------------------------
// Kernel 1: h[N,64] = x[N,128] @ W[64,128]^T  via V_WMMA_F32_16X16X4_F32.
// One wave per 16x16 output tile; K loop of 32 WMMA ops.
// A frag (16x4 f32, 2 VGPRs): lane L holds A[M=L%16][K = (L/16)*2 + v]
// B frag (4x16 f32, 2 VGPRs): lane L holds B[K = (L/16)*2 + v][N = L%16]
//   with B[k][n] = W[n][k]  -> same float2 addressing pattern as A.
// D/C (16x16 f32, 8 VGPRs): VGPR v, lane L -> row (v + (L/16)*8), col L%16.
// ---------------------------------------------------------------------------
__global__ __launch_bounds__(256)
void gat_gemm_wmma(const float* __restrict__ x, const float* __restrict__ W,
                   float* __restrict__ h) {
    const int wave = threadIdx.x >> 5;               // 0..7
    const int lane = threadIdx.x & 31;
    const int tile = blockIdx.x * 8 + wave;          // 2048 tiles total
    const int tm = (tile >> 2) * 16;                 // node-row tile (512 tiles)
    const int tn = (tile & 3) * 16;                  // out-feature tile (4 tiles)
    const int half = lane >> 4;                      // 0 or 1
    const int l16  = lane & 15;

    const float* arow = x + (size_t)(tm + l16) * IN_DIM;
    const float* brow = W + (size_t)(tn + l16) * IN_DIM;

    v8f c = {};
#pragma unroll
    for (int k0 = 0; k0 < IN_DIM; k0 += 4) {
        v2f a = *(const v2f*)(arow + k0 + half * 2);
        v2f b = *(const v2f*)(brow + k0 + half * 2);
        c = __builtin_amdgcn_wmma_f32_16x16x4_f32(
                /*neg_a=*/false, a, /*neg_b=*/false, b,
                /*c_mod=*/(short)0, c, /*reuse_a=*/false, /*reuse_b=*/false);
    }
#pragma unroll
    for (int v = 0; v < 8; ++v)
        h[(size_t)(tm + v + half * 8) * OUT_DIM + tn + l16] = c[v];
}

// ---------------------------------------------------------------------------
// Kernel 2: left[i] = h[i] . a[0:64], right[i] = h[i] . a[64:128]
// ---------------------------------------------------------------------------
__global__ __launch_bounds__(256)
void gat_node_dots(const float* __restrict__ h, const float* __restrict__ a,
                   float* __restrict__ left, float* __restrict__ right) {
    const int i = blockIdx.x * 256 + threadIdx.x;
    if (i >= N_NODES) return;
    const float4* hv = (const float4*)(h + (size_t)i * OUT_DIM);
    const float4* al = (const float4*)(a);
    const float4* ar = (const float4*)(a + OUT_DIM);
    float l = 0.f, r = 0.f;
#pragma unroll
    for (int k = 0; k < OUT_DIM / 4; ++k) {
        float4 hh = hv[k], aa = al[k], bb = ar[k];
        l += hh.x * aa.x + hh.y * aa.y + hh.z * aa.z + hh.w * aa.w;
        r += hh.x * bb.x + hh.y * bb.y + hh.z * bb.z + hh.w * bb.w;
    }
    left[i] = l;
    right[i] = r;
}

// ---------------------------------------------------------------------------
// Kernel 3: zero the adjacency bitmap (deterministic every launch).
// 2048 blocks x 256 threads x uint4 = exactly 2,097,152 words.
// ---------------------------------------------------------------------------
__global__ __launch_bounds__(256)
void gat_zero_bitmap(uint4* __restrict__ bmp) {
    const size_t idx = (size_t)blockIdx.x * 256 + threadIdx.x;
    bmp[idx] = make_uint4(0u, 0u, 0u, 0u);
}

// ---------------------------------------------------------------------------
// Kernel 4: scatter edges into bitmap; atomicOr naturally dedupes duplicates,
// matching the reference's dense-mask semantics.
// ---------------------------------------------------------------------------
__global__ __launch_bounds__(256)
void gat_scatter_edges(const int* __restrict__ edge_index,
                       unsigned* __restrict__ bmp) {
    const int e = blockIdx.x * 256 + threadIdx.x;
    if (e >= N_EDGES) return;
    const int r = edge_index[e];              // edge_index[0, e]
    const int c = edge_index[N_EDGES + e];    // edge_index[1, e]
    atomicOr(&bmp[(size_t)r * BMP_WORDS_PER_ROW + (c >> 5)], 1u << (c & 31));
}

// ---------------------------------------------------------------------------
// Kernel 5: one block (256 threads) per row:
//   scan 256 bitmap words -> compact neighbor list in LDS ->
//   stable softmax (max, sum-exp reductions) -> 4-way channel aggregation ->
//   elu epilogue.
// ---------------------------------------------------------------------------
#define NMAX 320   // far above max plausible degree for Poisson(32)

__global__ __launch_bounds__(256)
void gat_aggregate(const unsigned* __restrict__ bmp,
                   const float* __restrict__ h,
                   const float* __restrict__ left,
                   const float* __restrict__ right,
                   float* __restrict__ out) {
    __shared__ int   s_j[NMAX];
    __shared__ float s_v[NMAX];
    __shared__ float red[256];
    __shared__ int   s_cnt;

    const int row = blockIdx.x;
    const int tid = threadIdx.x;
    if (tid == 0) s_cnt = 0;
    __syncthreads();

    const float lrow = left[row];

    // --- build neighbor list + local max of leaky_relu(left+right) ---
    unsigned wbits = bmp[(size_t)row * BMP_WORDS_PER_ROW + tid];
    float lmax = -INFINITY;
    while (wbits) {
        const int b = __ffs(wbits) - 1;
        wbits &= wbits - 1;
        const int j = tid * 32 + b;
        const float s = lrow + right[j];
        const float v = (s > 0.f) ? s : NEG_SLOPE * s;   // leaky_relu
        lmax = fmaxf(lmax, v);
        const int pos = atomicAdd(&s_cnt, 1);
        if (pos < NMAX) { s_j[pos] = j; s_v[pos] = v; }
    }
    red[tid] = lmax;
    __syncthreads();
#pragma unroll
    for (int s = 128; s > 0; s >>= 1) {
        if (tid < s) red[tid] = fmaxf(red[tid], red[tid + s]);
        __syncthreads();
    }
    const float M = red[0];
    __syncthreads();

    const int cnt = min(s_cnt, NMAX);

    // --- sum of exp(v - M); overwrite s_v with exp values ---
    float lsum = 0.f;
    for (int k = tid; k < cnt; k += 256) {
        const float e = expf(s_v[k] - M);
        s_v[k] = e;
        lsum += e;
    }
    red[tid] = lsum;
    __syncthreads();
#pragma unroll
    for (int s = 128; s > 0; s >>= 1) {
        if (tid < s) red[tid] += red[tid + s];
        __syncthreads();
    }
    const float invS = 1.0f / red[0];
    __syncthreads();

    // --- weighted aggregation: 4 groups x 64 channels ---
    const int c = tid & 63;
    const int g = tid >> 6;
    float acc = 0.f;
    for (int k = g; k < cnt; k += 4)
        acc += s_v[k] * h[(size_t)s_j[k] * OUT_DIM + c];
    red[tid] = acc;
    __syncthreads();
    if (tid < 64) {
        float s = (red[c] + red[c + 64]) + (red[c + 128] + red[c + 192]);
        s *= invS;
        out[(size_t)row * OUT_DIM + c] = (s > 0.f) ? s : (expf(s) - 1.0f); // elu
    }
}

// ---------------------------------------------------------------------------
// Launch
// ---------------------------------------------------------------------------
extern "C" void kernel_launch(void* const* d_in, const int* in_sizes, int n_in,
                              void* d_out, int out_size, void* d_ws, size_t ws_size,
                              hipStream_t stream) {
    (void)in_sizes; (void)n_in; (void)out_size; (void)ws_size;
    const float* x  = (const float*)d_in[0];   // [N, 128]
    const int*   ei = (const int*)d_in[1];     // [2, E]
    const float* W  = (const float*)d_in[2];   // [64, 128]
    const float* a  = (const float*)d_in[3];   // [128]
    float* out = (float*)d_out;                // [N, 64]

    char* ws = (char*)d_ws;
    float*    h      = (float*)(ws + WS_H_OFF);
    float*    left   = (float*)(ws + WS_LEFT_OFF);
    float*    right  = (float*)(ws + WS_RIGHT_OFF);
    unsigned* bitmap = (unsigned*)(ws + WS_BMP_OFF);

    // h = x @ W^T : 2048 wave-tiles, 8 waves/block
    gat_gemm_wmma<<<256, 256, 0, stream>>>(x, W, h);
    // left/right node dots
    gat_node_dots<<<N_NODES / 256, 256, 0, stream>>>(h, a, left, right);
    // adjacency bitmap: zero then scatter
    gat_zero_bitmap<<<BMP_TOTAL_WORDS / (256 * 4), 256, 0, stream>>>((uint4*)bitmap);
    gat_scatter_edges<<<N_EDGES / 256, 256, 0, stream>>>(ei, bitmap);
    // softmax + aggregation + elu, one block per row
    gat_aggregate<<<N_NODES, 256, 0, stream>>>(bitmap, h, left, right, out);
}